// QValueNetBlock_35699768164382
// MI455X (gfx1250) — compile-verified
//
#include <hip/hip_runtime.h>

typedef __attribute__((ext_vector_type(2))) float v2f;
typedef __attribute__((ext_vector_type(8))) float v8f;
typedef unsigned int u32x4 __attribute__((ext_vector_type(4)));
typedef int i32x8 __attribute__((ext_vector_type(8)));
typedef int i32x4 __attribute__((ext_vector_type(4)));

#define D_DIM 512
#define BK 64
#define TILE_M 64          // rows per workgroup
#define TILE_N 64          // cols per workgroup (8 waves as 4 row-tiles x 2 col-tiles, 16x32 per wave)
#define LDS_STRIDE 68      // 272B row stride: 16B-aligned, conflict-free b64 fragment reads

// LDS byte offset of a __shared__ object (ptrtoint on the AS(3) pointer).
__device__ __forceinline__ unsigned lds_offset(const void* p) {
    return (unsigned)(unsigned long long)(__attribute__((address_space(3))) const void*)p;
}

// TDM: DMA a 64x64 f32 tile (row stride 512 floats) from global into LDS with
// hardware row padding of 4 DWORDs per 64 DWORDs -> LDS row stride 68 floats.
__device__ __forceinline__ void tdm_load_tile_64x64(const float* gptr, unsigned lds_byte_addr) {
    unsigned long long ga = (unsigned long long)gptr;   // byte address of tile start
    u32x4 g0;
    g0[0] = 1u;                                         // count=1, user descriptor
    g0[1] = lds_byte_addr;                              // lds_addr
    g0[2] = (unsigned)(ga & 0xFFFFFFFFull);             // global_addr[31:0]
    g0[3] = (unsigned)((ga >> 32) & 0x1FFFFFFull)       // global_addr[56:32]
          | (2u << 30);                                 // type=2 ("image")
    i32x8 g1;
    g1[0] = (2 << 16)        // data_size = 2 -> 4 bytes
          | (1 << 20)        // pad_enable
          | (5 << 22)        // pad_interval = 5 -> every 64 DWORDs
          | (3 << 25);       // pad_amount  = 3 -> 4 DWORDs
    g1[1] = (512 & 0xFFFF) << 16;        // tensor_dim0[15:0] = 512 (bits 63:48)
    g1[2] = (512 & 0xFFFF) << 16;        // tensor_dim0 hi = 0 | tensor_dim1[15:0] = 512
    g1[3] = (64 << 16);                  // tensor_dim1 hi = 0 | tile_dim0 = 64
    g1[4] = 64;                          // tile_dim1 = 64, tile_dim2 = 0
    g1[5] = 512;                         // tensor_dim0_stride[31:0] = 512 elements
    g1[6] = 0;                           // stride hi / dim1_stride lo (unused, 2D tile)
    g1[7] = 0;
    i32x4 gz4 = {0, 0, 0, 0};            // groups 2/3: unused for 2D tile
    i32x8 gz8 = {0, 0, 0, 0, 0, 0, 0, 0};
    // 6-arg form (clang-23 / therock-10.0 lane)
    __builtin_amdgcn_tensor_load_to_lds(g0, g1, gz4, gz4, gz8, 0);
}

// C[n,e] = sum_d A[n,d] * W[e,d] (+ bias[e]); optional fused prologue A' = A + states*w0
// and fused leaky_relu (slope 0.2) epilogue.
template <bool FUSE>
__global__ __launch_bounds__(256)
void gemm_wmma_f32(const float* __restrict__ A,
                   const float* __restrict__ W,      // [512,512] row-major (out-dim major)
                   const float* __restrict__ bias,   // [512]
                   const float* __restrict__ states, // [N] (FUSE only)
                   const float* __restrict__ w0p,    // scalar (FUSE only)
                   float* __restrict__ Out,          // [N,512]
                   int Nrows)
{
    __shared__ float As[TILE_M * LDS_STRIDE];
    __shared__ float Ws[TILE_N * LDS_STRIDE];

    const int tid  = threadIdx.x;
    const int wave = tid >> 5;          // 8 waves (wave32)
    const int lane = tid & 31;
    const int ln   = lane & 15;
    const int half = lane >> 4;         // K-half select per ISA f32 A/B fragment layout
    const int wm   = wave & 3;          // 4 row tiles of 16
    const int wn   = wave >> 2;         // 2 col tiles of 32

    const int row0 = blockIdx.x * TILE_M;
    const int col0 = blockIdx.y * TILE_N;

    const float w0 = FUSE ? w0p[0] : 0.0f;
    const unsigned ws_lds = lds_offset(&Ws[0]);

    v8f acc0 = {};
    v8f acc1 = {};

    for (int kk = 0; kk < D_DIM; kk += BK) {
        // ---- W tile via Tensor Data Mover: 64 (e) x 64 (k), padded to stride 68 ----
        if (wave == 0) {
            tdm_load_tile_64x64(W + (size_t)col0 * D_DIM + kk, ws_lds);
        }
        // ---- A tile staged by all threads (fused prologue needs VALU anyway) ----
        #pragma unroll
        for (int p = 0; p < 4; ++p) {
            int idx = p * 256 + tid;          // 0..1023 float4 slots over 64x64
            int r   = idx >> 4;
            int c4  = (idx & 15) << 2;
            int gr  = row0 + r;
            float4 v = make_float4(0.f, 0.f, 0.f, 0.f);
            if (gr < Nrows) {
                v = *(const float4*)(A + (size_t)gr * D_DIM + kk + c4);
                if (FUSE) {
                    float s = states[gr] * w0;
                    v.x += s; v.y += s; v.z += s; v.w += s;
                }
            }
            *(float4*)(&As[r * LDS_STRIDE + c4]) = v;
        }
        if (wave == 0) {
            __builtin_amdgcn_s_wait_tensorcnt(0);   // TDM data visible in LDS
        }
        __syncthreads();

        // ---- 32 x v_wmma_f32_16x16x4_f32 per K-chunk: 2 independent acc chains ----
        // A frag (16x4): lane ln = row M, vgpr pair = K = 2*half + {0,1}
        // B frag (4x16): lane ln = col N, vgpr pair = K = 2*half + {0,1}; B[k][n] = W[e0+n][k]
        const float* ab  = &As[(wm * 16 + ln) * LDS_STRIDE + 2 * half];
        const float* bb0 = &Ws[(wn * 32 + ln) * LDS_STRIDE + 2 * half];
        const float* bb1 = &Ws[(wn * 32 + 16 + ln) * LDS_STRIDE + 2 * half];
        #pragma unroll
        for (int k = 0; k < BK; k += 4) {
            v2f a  = *(const v2f*)(ab + k);
            v2f b0 = *(const v2f*)(bb0 + k);
            v2f b1 = *(const v2f*)(bb1 + k);
            acc0 = __builtin_amdgcn_wmma_f32_16x16x4_f32(false, a, false, b0,
                                                         (short)0, acc0, false, false);
            acc1 = __builtin_amdgcn_wmma_f32_16x16x4_f32(false, a, false, b1,
                                                         (short)0, acc1, false, false);
        }
        __syncthreads();
    }

    // ---- epilogue: bias (+ leaky relu), store per ISA C/D layout ----
    const int   n0  = col0 + wn * 32 + ln;
    const int   n1  = n0 + 16;
    const float bn0 = bias[n0];
    const float bn1 = bias[n1];
    #pragma unroll
    for (int v = 0; v < 8; ++v) {
        int m = row0 + wm * 16 + v + 8 * half;
        if (m < Nrows) {
            float v0 = acc0[v] + bn0;
            float v1 = acc1[v] + bn1;
            if (FUSE) {
                v0 = (v0 > 0.f) ? v0 : 0.2f * v0;
                v1 = (v1 > 0.f) ? v1 : 0.2f * v1;
            }
            Out[(size_t)m * D_DIM + n0] = v0;
            Out[(size_t)m * D_DIM + n1] = v1;
        }
    }
}

__global__ void zero_f4(float* __restrict__ out, int n4) {
    int i = blockIdx.x * blockDim.x + threadIdx.x;
    if (i < n4) ((float4*)out)[i] = make_float4(0.f, 0.f, 0.f, 0.f);
}

// One workgroup per edge: out[dst] += ew * h[src]; h & out are L2-resident (102 MB each,
// 192 MB L2), so gather + f32 atomics are L2-bandwidth bound.
__global__ __launch_bounds__(256)
void edge_scatter_add(const float* __restrict__ h,
                      const float* __restrict__ ew,
                      const int*   __restrict__ ei,   // [2*E] flat: src rows then dst rows
                      float* __restrict__ out,
                      int E)
{
    int e   = blockIdx.x;
    int src = ei[e];
    int dst = ei[E + e];
    float w = ew[e];
    const float2* hr = (const float2*)(h + (size_t)src * D_DIM);
    float* orow = out + (size_t)dst * D_DIM;
    int t = threadIdx.x;                 // 256 threads x 2 floats = 512
    float2 v = hr[t];
    atomicAdd(&orow[2 * t],     w * v.x);
    atomicAdd(&orow[2 * t + 1], w * v.y);
}

extern "C" void kernel_launch(void* const* d_in, const int* in_sizes, int n_in,
                              void* d_out, int out_size, void* d_ws, size_t ws_size,
                              hipStream_t stream) {
    (void)in_sizes; (void)n_in; (void)out_size; (void)ws_size;
    const float* x      = (const float*)d_in[0];
    const float* ew     = (const float*)d_in[1];
    const float* states = (const float*)d_in[2];
    const float* w0     = (const float*)d_in[3];
    const float* w1     = (const float*)d_in[4];
    const float* b1     = (const float*)d_in[5];
    const float* w2     = (const float*)d_in[6];
    const float* b2     = (const float*)d_in[7];
    const int*   ei     = (const int*)d_in[8];
    float* out = (float*)d_out;

    const int N = 50000;
    const int E = 400000;

    float* h1 = (float*)d_ws;
    float* h2 = h1 + (size_t)N * D_DIM;

    dim3 grid((N + TILE_M - 1) / TILE_M, D_DIM / TILE_N);   // 782 x 8
    gemm_wmma_f32<true><<<grid, 256, 0, stream>>>(x,  w1, b1, states, w0, h1, N);
    gemm_wmma_f32<false><<<grid, 256, 0, stream>>>(h1, w2, b2, nullptr, nullptr, h2, N);

    int n4 = (N * D_DIM) / 4;
    zero_f4<<<(n4 + 255) / 256, 256, 0, stream>>>(out, n4);

    edge_scatter_add<<<E, 256, 0, stream>>>(h2, ew, ei, out, E);
}